// ASTGCN_noSatt_82867099009465
// MI455X (gfx1250) — compile-verified
//
#include <hip/hip_runtime.h>
#include <math.h>

// ---------------------------------------------------------------------------
// ASTGCN (no spatial attention) forward for MI455X (gfx1250, wave32, WMMA)
// Dominant work: dense 3000x3000 graph GEMMs -> V_WMMA_F32_16X16X4_F32
// Staging: GLOBAL_LOAD_ASYNC_TO_LDS_B128 (ASYNCcnt) for full tiles,
//          guarded manual copy (with zero-fill) for edge tiles.
// ---------------------------------------------------------------------------

typedef __attribute__((ext_vector_type(2))) float v2f;
typedef __attribute__((ext_vector_type(8))) float v8f;

#define GN 3000      // graph nodes
#define GB 8         // batch
#define T2A 64
#define T2B 32
#define TP 12

#define BM 128
#define BN 64
#define BK 32
#define LDA (BK + 4)   // row stride 36 floats = 144B (16B aligned rows)
#define LDB (BN + 4)   // row stride 68 floats = 272B (16B aligned rows)

// ---- CDNA5 async global->LDS copy (tracked by ASYNCcnt) -------------------
__device__ __forceinline__ void async_g2l_b128(void* lds_ptr, const void* gptr)
{
#if defined(__HIP_DEVICE_COMPILE__)
    unsigned loff = (unsigned)(uintptr_t)lds_ptr;   // low 32 bits = LDS offset
    asm volatile("global_load_async_to_lds_b128 %0, %1, off"
                 :: "v"(loff), "v"(gptr) : "memory");
#endif
}

__device__ __forceinline__ void wait_async0()
{
#if defined(__HIP_DEVICE_COMPILE__)
    asm volatile("s_wait_asynccnt 0x0" ::: "memory");
#endif
}

// ---------------------------------------------------------------------------
// Generic f32 WMMA GEMM:  Y = alpha * (A @ B) [+ beta * Z] [+ bias(col)] [relu]
// A: (M x K) row-major, B: (K x Cn) row-major, Y/Z: (M x Cn) row-major
// Block: 256 threads = 8 waves; tile 128x64; each wave computes 16 rows x 64 cols.
// ---------------------------------------------------------------------------
__global__ __launch_bounds__(256)
void gemm_f32_wmma(const float* __restrict__ A, const float* __restrict__ Bm,
                   const float* __restrict__ Z, const float* __restrict__ bias,
                   float* __restrict__ Y,
                   int M, int K, int Cn,
                   float alpha, float beta, int relu)
{
    __shared__ float As[BM * LDA];
    __shared__ float Bs[BK * LDB];

    const int tid  = threadIdx.x;
    const int lane = tid & 31;
    const int wv   = tid >> 5;       // wave 0..7
    const int hl   = lane >> 4;      // lane half
    const int l16  = lane & 15;

    const int rowBase = blockIdx.y * BM;
    const int colBase = blockIdx.x * BN;
    const int arow    = wv * 16 + l16;

    v8f acc[4];
#pragma unroll
    for (int j = 0; j < 4; ++j)
#pragma unroll
        for (int e = 0; e < 8; ++e) acc[j][e] = 0.0f;

    for (int k0 = 0; k0 < K; k0 += BK) {
        const bool fullK = (k0 + BK) <= K;
        const bool fullA = fullK && ((rowBase + BM) <= M);
        const bool fullB = fullK && ((colBase + BN) <= Cn);

        // prefetch next A chunk row into L2 (global_prefetch_b8)
        if (k0 + BK < K) {
            int pr = rowBase + (tid & 127);
            if (pr < M)
                __builtin_prefetch(A + (size_t)pr * K + (k0 + BK), 0, 1);
        }

        // ---- stage A tile: 128 rows x 32 k
        if (fullA) {
#pragma unroll
            for (int i = 0; i < 4; ++i) {
                int idx = tid + i * 256;
                int r  = idx >> 3;
                int kq = (idx & 7) << 2;
                async_g2l_b128(&As[r * LDA + kq],
                               A + (size_t)(rowBase + r) * K + (k0 + kq));
            }
        } else {
#pragma unroll
            for (int i = 0; i < 4; ++i) {
                int idx = tid + i * 256;
                int r  = idx >> 3;
                int kq = (idx & 7) << 2;
                int gr = rowBase + r, gk = k0 + kq;
                float4 v = make_float4(0.f, 0.f, 0.f, 0.f);
                if (gr < M) {
                    if (gk + 3 < K) {
                        v = *(const float4*)(A + (size_t)gr * K + gk);
                    } else {
                        if (gk + 0 < K) v.x = A[(size_t)gr * K + gk + 0];
                        if (gk + 1 < K) v.y = A[(size_t)gr * K + gk + 1];
                        if (gk + 2 < K) v.z = A[(size_t)gr * K + gk + 2];
                        if (gk + 3 < K) v.w = A[(size_t)gr * K + gk + 3];
                    }
                }
                *(float4*)(&As[r * LDA + kq]) = v;
            }
        }

        // ---- stage B tile: 32 k x 64 cols
        if (fullB) {
#pragma unroll
            for (int i = 0; i < 2; ++i) {
                int idx = tid + i * 256;
                int r  = idx >> 4;
                int cq = (idx & 15) << 2;
                async_g2l_b128(&Bs[r * LDB + cq],
                               Bm + (size_t)(k0 + r) * Cn + (colBase + cq));
            }
        } else {
#pragma unroll
            for (int i = 0; i < 2; ++i) {
                int idx = tid + i * 256;
                int r  = idx >> 4;
                int cq = (idx & 15) << 2;
                int gk = k0 + r, gc = colBase + cq;
                float4 v = make_float4(0.f, 0.f, 0.f, 0.f);
                if (gk < K) {
                    if (gc + 3 < Cn) {
                        v = *(const float4*)(Bm + (size_t)gk * Cn + gc);
                    } else {
                        if (gc + 0 < Cn) v.x = Bm[(size_t)gk * Cn + gc + 0];
                        if (gc + 1 < Cn) v.y = Bm[(size_t)gk * Cn + gc + 1];
                        if (gc + 2 < Cn) v.z = Bm[(size_t)gk * Cn + gc + 2];
                        if (gc + 3 < Cn) v.w = Bm[(size_t)gk * Cn + gc + 3];
                    }
                }
                *(float4*)(&Bs[r * LDB + cq]) = v;
            }
        }

        wait_async0();       // ASYNCcnt==0 -> async tile data landed in LDS
        __syncthreads();

        // ---- compute: 8 k-steps of 4, software-pipelined fragment loads
        {
            v2f af[2];
            v2f bf[2][4];
            // preload k-step 0
            {
                const int kb = 0 * 4 + 2 * hl;
                af[0].x = As[arow * LDA + kb + 0];
                af[0].y = As[arow * LDA + kb + 1];
#pragma unroll
                for (int j = 0; j < 4; ++j) {
                    bf[0][j].x = Bs[(kb + 0) * LDB + j * 16 + l16];
                    bf[0][j].y = Bs[(kb + 1) * LDB + j * 16 + l16];
                }
            }
#pragma unroll
            for (int kk = 0; kk < BK / 4; ++kk) {
                const int cur = kk & 1;
                const int nxt = cur ^ 1;
                if (kk + 1 < BK / 4) {
                    const int kb = (kk + 1) * 4 + 2 * hl;
                    af[nxt].x = As[arow * LDA + kb + 0];
                    af[nxt].y = As[arow * LDA + kb + 1];
#pragma unroll
                    for (int j = 0; j < 4; ++j) {
                        bf[nxt][j].x = Bs[(kb + 0) * LDB + j * 16 + l16];
                        bf[nxt][j].y = Bs[(kb + 1) * LDB + j * 16 + l16];
                    }
                }
#pragma unroll
                for (int j = 0; j < 4; ++j) {
                    acc[j] = __builtin_amdgcn_wmma_f32_16x16x4_f32(
                        false, af[cur], false, bf[cur][j], (short)0, acc[j],
                        false, false);
                }
            }
        }
        __syncthreads();
    }

    // ---- epilogue: lane holds rows (v + 8*hl), col = colBase + j*16 + l16
#pragma unroll
    for (int j = 0; j < 4; ++j) {
        int col = colBase + j * 16 + l16;
        if (col < Cn) {
#pragma unroll
            for (int v = 0; v < 8; ++v) {
                int row = rowBase + wv * 16 + v + 8 * hl;
                if (row < M) {
                    float val = alpha * acc[j][v];
                    if (Z)    val += beta * Z[(size_t)row * Cn + col];
                    if (bias) val += bias[col];
                    if (relu) val = fmaxf(val, 0.0f);
                    Y[(size_t)row * Cn + col] = val;
                }
            }
        }
    }
}

// ---------------------------------------------------------------------------
// Laplacian: dinv[n] = deg>0 ? deg^-0.5 : 0 (deg = off-diagonal row sum of A)
// ---------------------------------------------------------------------------
__global__ void deg_kernel(const float* __restrict__ A, float* __restrict__ dinv, int n)
{
    __shared__ float red[256];
    int row = blockIdx.x;
    float s = 0.f;
    for (int m = threadIdx.x; m < n; m += 256)
        if (m != row) s += A[(size_t)row * n + m];
    red[threadIdx.x] = s;
    __syncthreads();
    for (int off = 128; off > 0; off >>= 1) {
        if (threadIdx.x < off) red[threadIdx.x] += red[threadIdx.x + off];
        __syncthreads();
    }
    if (threadIdx.x == 0) {
        float d = red[0];
        dinv[row] = (d > 0.f) ? (1.0f / sqrtf(d)) : 0.f;
    }
}

__global__ void build_L(const float* __restrict__ A, const float* __restrict__ dinv,
                        float* __restrict__ Lout, int n)
{
    int idx = blockIdx.x * blockDim.x + threadIdx.x;
    int total = n * n;
    if (idx >= total) return;
    int row = idx / n, col = idx - row * n;
    Lout[idx] = (row == col) ? 0.f : -(dinv[row] * A[idx] * dinv[col]);
}

// ---------------------------------------------------------------------------
// Pack X (B,N,1,T) -> Xp (N, B, T)
// ---------------------------------------------------------------------------
__global__ void pack_kernel(const float* __restrict__ X, float* __restrict__ Xp,
                            int n, int T)
{
    int idx = blockIdx.x * blockDim.x + threadIdx.x;
    int total = n * GB * T;
    if (idx >= total) return;
    int nn  = idx / (GB * T);
    int rem = idx - nn * (GB * T);
    int b = rem / T, t = rem - b * T;
    Xp[idx] = X[((size_t)b * n + nn) * T + t];
}

// ---------------------------------------------------------------------------
// tconv (Conv1d k=3 pad=1 over t) + ReLU.  G,H: (NB, T)
// ---------------------------------------------------------------------------
__global__ void tconv_relu(const float* __restrict__ G, float* __restrict__ H,
                           int NB, int T,
                           const float* __restrict__ cw, const float* __restrict__ cb)
{
    int idx = blockIdx.x * blockDim.x + threadIdx.x;
    int total = NB * T;
    if (idx >= total) return;
    int r = idx / T, t = idx - r * T;
    float xm = (t > 0)     ? G[(size_t)r * T + t - 1] : 0.f;
    float x0 = G[idx];
    float xp = (t < T - 1) ? G[(size_t)r * T + t + 1] : 0.f;
    H[idx] = fmaxf(cw[0] * xm + cw[1] * x0 + cw[2] * xp + cb[0], 0.f);
}

// ---------------------------------------------------------------------------
// Final linear + ReLU + fusion accumulate into out (B,N,1,Tp)
// H: (N,B,T2B); mode 0: out = F*y ; mode 1: out += F*y
// ---------------------------------------------------------------------------
__global__ void final_fuse(const float* __restrict__ H, const float* __restrict__ WD,
                           const float* __restrict__ bD, const float* __restrict__ F,
                           float* __restrict__ out, int n, int mode)
{
    int id = blockIdx.x * blockDim.x + threadIdx.x;
    if (id >= n * GB) return;
    int nn = id / GB, b = id - nn * GB;
    const float* h = H + (size_t)id * T2B;
#pragma unroll
    for (int p = 0; p < TP; ++p) {
        float s = bD[p];
        for (int t = 0; t < T2B; ++t) s += h[t] * WD[t * TP + p];
        s = fmaxf(s, 0.f) * F[p];
        float* o = out + ((size_t)b * n + nn) * TP + p;
        if (mode == 0) *o = s; else *o += s;
    }
}

// ---------------------------------------------------------------------------
// Host-side orchestration
// ---------------------------------------------------------------------------
static void launch_gemm(const float* A, const float* Bm, const float* Z,
                        const float* bias, float* Y,
                        int M, int K, int Cn, float alpha, float beta, int relu,
                        hipStream_t s)
{
    dim3 grid((Cn + BN - 1) / BN, (M + BM - 1) / BM);
    gemm_f32_wmma<<<grid, 256, 0, s>>>(A, Bm, Z, bias, Y, M, K, Cn, alpha, beta, relu);
}

static void run_branch(const float* X, int T,
                       const float* W1, const float* bg1, const float* c1w, const float* c1b,
                       const float* W2, const float* bg2, const float* c2w, const float* c2b,
                       const float* WD, const float* bD, const float* F, int mode,
                       const float* L, float* b0, float* b1, float* b2, float* b3,
                       float* out, hipStream_t s)
{
    const int NB = GN * GB;
    // pack input -> b0 (Tx0, layer 1)
    {
        int total = GN * GB * T;
        pack_kernel<<<(total + 255) / 256, 256, 0, s>>>(X, b0, GN, T);
    }
    int C1 = GB * T;
    // layer 1 Cheb: Tx1 = L@Tx0 ; Tx2 = 2*L@Tx1 - Tx0
    launch_gemm(L, b0, nullptr, nullptr, b1, GN, GN, C1, 1.f, 0.f, 0, s);
    launch_gemm(L, b1, b0, nullptr, b2, GN, GN, C1, 2.f, -1.f, 0, s);
    // G = relu(Tx0@W0 + Tx1@W1 + Tx2@W2 + b)
    launch_gemm(b0, W1 + 0 * T * T2A, nullptr, nullptr, b3, NB, T, T2A, 1.f, 0.f, 0, s);
    launch_gemm(b1, W1 + 1 * T * T2A, b3, nullptr, b3, NB, T, T2A, 1.f, 1.f, 0, s);
    launch_gemm(b2, W1 + 2 * T * T2A, b3, bg1, b3, NB, T, T2A, 1.f, 1.f, 1, s);
    // H = relu(tconv(G)) -> b0 (becomes Tx0 of layer 2)
    tconv_relu<<<(NB * T2A + 255) / 256, 256, 0, s>>>(b3, b0, NB, T2A, c1w, c1b);

    int C2 = GB * T2A;
    launch_gemm(L, b0, nullptr, nullptr, b1, GN, GN, C2, 1.f, 0.f, 0, s);
    launch_gemm(L, b1, b0, nullptr, b2, GN, GN, C2, 2.f, -1.f, 0, s);
    launch_gemm(b0, W2 + 0 * T2A * T2B, nullptr, nullptr, b3, NB, T2A, T2B, 1.f, 0.f, 0, s);
    launch_gemm(b1, W2 + 1 * T2A * T2B, b3, nullptr, b3, NB, T2A, T2B, 1.f, 1.f, 0, s);
    launch_gemm(b2, W2 + 2 * T2A * T2B, b3, bg2, b3, NB, T2A, T2B, 1.f, 1.f, 1, s);
    tconv_relu<<<(NB * T2B + 255) / 256, 256, 0, s>>>(b3, b0, NB, T2B, c2w, c2b);

    final_fuse<<<(NB + 255) / 256, 256, 0, s>>>(b0, WD, bD, F, out, GN, mode);
}

extern "C" void kernel_launch(void* const* d_in, const int* in_sizes, int n_in,
                              void* d_out, int out_size, void* d_ws, size_t ws_size,
                              hipStream_t stream)
{
    const float* Xh  = (const float*)d_in[0];
    const float* Xd  = (const float*)d_in[1];
    const float* Xw  = (const float*)d_in[2];
    const float* A   = (const float*)d_in[3];
    const float* Wh1 = (const float*)d_in[4];  const float* bh1 = (const float*)d_in[5];
    const float* ch1w= (const float*)d_in[6];  const float* ch1b= (const float*)d_in[7];
    const float* Wh2 = (const float*)d_in[8];  const float* bh2 = (const float*)d_in[9];
    const float* ch2w= (const float*)d_in[10]; const float* ch2b= (const float*)d_in[11];
    const float* Wd1 = (const float*)d_in[12]; const float* bd1 = (const float*)d_in[13];
    const float* cd1w= (const float*)d_in[14]; const float* cd1b= (const float*)d_in[15];
    const float* Wd2 = (const float*)d_in[16]; const float* bd2 = (const float*)d_in[17];
    const float* cd2w= (const float*)d_in[18]; const float* cd2b= (const float*)d_in[19];
    const float* Ww1 = (const float*)d_in[20]; const float* bw1 = (const float*)d_in[21];
    const float* cw1w= (const float*)d_in[22]; const float* cw1b= (const float*)d_in[23];
    const float* Ww2 = (const float*)d_in[24]; const float* bw2 = (const float*)d_in[25];
    const float* cw2w= (const float*)d_in[26]; const float* cw2b= (const float*)d_in[27];
    const float* WD  = (const float*)d_in[28]; const float* bD  = (const float*)d_in[29];
    const float* Fh  = (const float*)d_in[30];
    const float* Fd  = (const float*)d_in[31];
    const float* Fw  = (const float*)d_in[32];

    float* out = (float*)d_out;
    float* ws  = (float*)d_ws;

    // workspace layout (floats, all 16B-aligned offsets)
    float* L    = ws;                         // 9,000,000
    float* dinv = ws + 9000000;               // 3072 (3000 used)
    float* b0   = ws + 9003072;               // 4 activation buffers of 3000*8*64
    float* b1   = b0 + 1536000;
    float* b2   = b1 + 1536000;
    float* b3   = b2 + 1536000;

    // Laplacian: L_hat = -D^{-1/2} A_offdiag D^{-1/2}
    deg_kernel<<<GN, 256, 0, stream>>>(A, dinv, GN);
    {
        int total = GN * GN;
        build_L<<<(total + 255) / 256, 256, 0, stream>>>(A, dinv, L, GN);
    }

    run_branch(Xh, 24, Wh1, bh1, ch1w, ch1b, Wh2, bh2, ch2w, ch2b,
               WD, bD, Fh, /*mode=*/0, L, b0, b1, b2, b3, out, stream);
    run_branch(Xd, 12, Wd1, bd1, cd1w, cd1b, Wd2, bd2, cd2w, cd2b,
               WD, bD, Fd, /*mode=*/1, L, b0, b1, b2, b3, out, stream);
    run_branch(Xw, 24, Ww1, bw1, cw1w, cw1b, Ww2, bw2, cw2w, cw2b,
               WD, bD, Fw, /*mode=*/1, L, b0, b1, b2, b3, out, stream);
}